// ConversationAwareSAGEConv_19413252177995
// MI455X (gfx1250) — compile-verified
//
#include <hip/hip_runtime.h>
#include <hip/hip_bf16.h>

// ConversationAwareSAGEConv for MI455X (gfx1250).
//
//  * per-edge MLP hoisted to per-node (10x GEMM reduction)
//  * dst-CSR build (int atomics) replaces 128M fp32 scatter atomics with
//    contention-free gathers; both aggregations fused in one pass, one wave
//    per node, float4 (b128) row gathers: 32 lanes x 16B = one 512B row
//  * all GEMMs in full-precision V_WMMA_F32_16X16X4_F32 (op is bandwidth
//    bound, ~10 GFLOP total; bf16 would gain nothing and lose accuracy)
//  * std matrix kept in LDS (padded, bank-conflict-free) and consumed by the
//    gate GEMM without a global round-trip

typedef float v2f __attribute__((ext_vector_type(2)));
typedef float v8f __attribute__((ext_vector_type(8)));

#define DDIM 128
#define EPS_LN 1e-5f

__device__ __forceinline__ v8f v8f_zero() {
  v8f z;
#pragma unroll
  for (int i = 0; i < 8; ++i) z[i] = 0.f;
  return z;
}

__device__ __forceinline__ v8f wmma_f32(v2f a, v2f b, v8f c) {
  // V_WMMA_F32_16X16X4_F32: D = A(16x4) * B(4x16) + C(16x16), full fp32.
  return __builtin_amdgcn_wmma_f32_16x16x4_f32(
      /*neg_a=*/false, a, /*neg_b=*/false, b,
      /*c_mod=*/(short)0, c, /*reuse_a=*/false, /*reuse_b=*/false);
}

// Accumulate a 16x128 strip: acc += Arow(16x128) * B(128x128).
// Arow = this lane's A-row base pointer (row lm of the strip, stride 128).
// Layout per ISA: lanes 0-15 carry K=k,k+1; lanes 16-31 carry K=k+2,k+3.
__device__ __forceinline__ void gemm_panel(const float* __restrict__ Arow,
                                           const float* __restrict__ B,
                                           int lm, int lh, v8f acc[8]) {
  for (int k = 0; k < DDIM; k += 4) {
    const int k0 = k + lh * 2;
    const float2 av = *(const float2*)(Arow + k0);
    v2f a;
    a.x = av.x;
    a.y = av.y;
#pragma unroll
    for (int t = 0; t < 8; ++t) {
      v2f b;
      b.x = B[(size_t)k0 * DDIM + t * 16 + lm];
      b.y = B[(size_t)(k0 + 1) * DDIM + t * 16 + lm];
      acc[t] = wmma_f32(a, b, acc[t]);
    }
  }
}

// ---------------------------------------------------------------- CSR build
__global__ void k_zero(int* __restrict__ cnt, int N) {
  int i = blockIdx.x * blockDim.x + threadIdx.x;
  if (i < N) cnt[i] = 0;
}

__global__ void k_count(const int* __restrict__ dst, int* __restrict__ cnt,
                        int E) {
  for (int i = blockIdx.x * blockDim.x + threadIdx.x; i < E;
       i += gridDim.x * blockDim.x)
    atomicAdd(&cnt[dst[i]], 1);
}

#define SCAN_B 1024
__global__ void k_scan1(const int* __restrict__ cnt, int* __restrict__ part,
                        int N) {
  __shared__ int sm[SCAN_B];
  int i = blockIdx.x * SCAN_B + threadIdx.x;
  sm[threadIdx.x] = (i < N) ? cnt[i] : 0;
  __syncthreads();
  for (int s = SCAN_B / 2; s > 0; s >>= 1) {
    if ((int)threadIdx.x < s) sm[threadIdx.x] += sm[threadIdx.x + s];
    __syncthreads();
  }
  if (threadIdx.x == 0) part[blockIdx.x] = sm[0];
}

__global__ void k_scan2(int* __restrict__ part, int nb,
                        int* __restrict__ rowptr_last) {
  if (threadIdx.x == 0 && blockIdx.x == 0) {
    int run = 0;
    for (int i = 0; i < nb; ++i) {
      int v = part[i];
      part[i] = run;
      run += v;
    }
    *rowptr_last = run;  // == E
  }
}

__global__ void k_scan3(const int* __restrict__ cnt,
                        const int* __restrict__ part, int* __restrict__ rowptr,
                        int* __restrict__ cursor, int N) {
  __shared__ int sm[SCAN_B];
  int i = blockIdx.x * SCAN_B + threadIdx.x;
  int x = (i < N) ? cnt[i] : 0;
  sm[threadIdx.x] = x;
  __syncthreads();
  for (int off = 1; off < SCAN_B; off <<= 1) {
    int v = ((int)threadIdx.x >= off) ? sm[threadIdx.x - off] : 0;
    __syncthreads();
    sm[threadIdx.x] += v;
    __syncthreads();
  }
  if (i < N) {
    int ex = part[blockIdx.x] + sm[threadIdx.x] - x;  // exclusive
    rowptr[i] = ex;
    cursor[i] = ex;
  }
}

__global__ void k_fill(const int* __restrict__ src, const int* __restrict__ dst,
                       int* __restrict__ cursor, int* __restrict__ esrc,
                       int E) {
  for (int i = blockIdx.x * blockDim.x + threadIdx.x; i < E;
       i += gridDim.x * blockDim.x) {
    int pos = atomicAdd(&cursor[dst[i]], 1);
    esrc[pos] = src[i];
  }
}

// ------------------------------------------------ per-node message MLP + LN
__global__ void __launch_bounds__(256) k_msg(
    const float* __restrict__ h, const float* __restrict__ ctx,
    const float* __restrict__ Wm, const float* __restrict__ bm,
    const float* __restrict__ lng, const float* __restrict__ lnb,
    float* __restrict__ msg, int N) {
  const int wave = threadIdx.x >> 5;
  const int strip = blockIdx.x * 8 + wave;
  const int nStrips = (N + 15) >> 4;
  if (strip >= nStrips) return;  // wave-uniform; EXEC stays all-1
  const int lane = threadIdx.x & 31;
  const int lm = lane & 15, lh = lane >> 4;
  const int rA = min(strip * 16 + lm, N - 1);

  v8f acc[8];
#pragma unroll
  for (int t = 0; t < 8; ++t) acc[t] = v8f_zero();

  gemm_panel(h + (size_t)rA * DDIM, Wm, lm, lh, acc);                 // h part
  gemm_panel(ctx + (size_t)rA * DDIM, Wm + DDIM * DDIM, lm, lh, acc); // ctx

#pragma unroll
  for (int t = 0; t < 8; ++t) {
    const float bb = bm[t * 16 + lm];
#pragma unroll
    for (int r = 0; r < 8; ++r) acc[t][r] += bb;
  }

  // LayerNorm: C-layout row (r + 8*lh) spans 8 accs x 16 lanes of this half.
  float mu[8], rsg[8];
#pragma unroll
  for (int r = 0; r < 8; ++r) {
    float s = 0.f;
#pragma unroll
    for (int t = 0; t < 8; ++t) s += acc[t][r];
    s += __shfl_xor(s, 1);
    s += __shfl_xor(s, 2);
    s += __shfl_xor(s, 4);
    s += __shfl_xor(s, 8);
    const float mean = s * (1.f / 128.f);
    float q = 0.f;
#pragma unroll
    for (int t = 0; t < 8; ++t) {
      const float d0 = acc[t][r] - mean;
      q += d0 * d0;
    }
    q += __shfl_xor(q, 1);
    q += __shfl_xor(q, 2);
    q += __shfl_xor(q, 4);
    q += __shfl_xor(q, 8);
    mu[r] = mean;
    rsg[r] = rsqrtf(q * (1.f / 128.f) + EPS_LN);
  }

#pragma unroll
  for (int t = 0; t < 8; ++t) {
    const int col = t * 16 + lm;
    const float g = lng[col], b = lnb[col];
#pragma unroll
    for (int r = 0; r < 8; ++r) {
      const int row = strip * 16 + r + lh * 8;
      if (row < N) {
        float v = (acc[t][r] - mu[r]) * rsg[r] * g + b;
        msg[(size_t)row * DDIM + col] = fmaxf(v, 0.f);
      }
    }
  }
}

// ------------------- fused gather-mean of h[src] and msg[src], 1 wave/node.
// Each lane owns 4 columns; one edge row = 32 lanes x float4 = 512B (b128).
__global__ void __launch_bounds__(256) k_agg(
    const float* __restrict__ h, const float* __restrict__ msg,
    const int* __restrict__ rowptr, const int* __restrict__ esrc,
    float* __restrict__ nmean, float* __restrict__ cagg, int N) {
  const int node = blockIdx.x * 8 + (threadIdx.x >> 5);
  if (node >= N) return;  // wave-uniform
  const int lane = threadIdx.x & 31;
  const int coff = lane * 4;
  const int s0 = rowptr[node], s1 = rowptr[node + 1];  // wave-uniform -> SMEM
  float4 sh = {0.f, 0.f, 0.f, 0.f};
  float4 sm = {0.f, 0.f, 0.f, 0.f};
  for (int i = s0; i < s1; ++i) {
    const int s = esrc[i];
    if (i + 1 < s1) {
      const int sn = esrc[i + 1];
      __builtin_prefetch(&h[(size_t)sn * DDIM + coff], 0, 0);
      __builtin_prefetch(&msg[(size_t)sn * DDIM + coff], 0, 0);
    }
    const float4 hv = *(const float4*)(h + (size_t)s * DDIM + coff);
    const float4 mv = *(const float4*)(msg + (size_t)s * DDIM + coff);
    sh.x += hv.x; sh.y += hv.y; sh.z += hv.z; sh.w += hv.w;
    sm.x += mv.x; sm.y += mv.y; sm.z += mv.z; sm.w += mv.w;
  }
  const int c = s1 - s0;
  const float inv = (c > 0) ? 1.f / (float)c : 0.f;  // c==0 -> sums are 0
  const float4 o1 = {sh.x * inv, sh.y * inv, sh.z * inv, sh.w * inv};
  const float4 o2 = {sm.x * inv, sm.y * inv, sm.z * inv, sm.w * inv};
  *(float4*)(nmean + (size_t)node * DDIM + coff) = o1;
  *(float4*)(cagg + (size_t)node * DDIM + coff) = o2;
}

// ---------------- std GEMM -> LDS -> gate GEMM -> fused blend, one pass
__global__ void __launch_bounds__(128) k_stdgate(
    const float* __restrict__ h, const float* __restrict__ nmean,
    const float* __restrict__ cagg, const float* __restrict__ Wself,
    const float* __restrict__ Wneigh, const float* __restrict__ bsage,
    const float* __restrict__ Wg, const float* __restrict__ bg,
    float* __restrict__ out, int N) {
  __shared__ float sstd[4][16 * 129];  // +1 pad: conflict-free A re-reads
  const int wave = threadIdx.x >> 5;
  const int strip = blockIdx.x * 4 + wave;
  const int nStrips = (N + 15) >> 4;
  if (strip >= nStrips) return;
  const int lane = threadIdx.x & 31;
  const int lm = lane & 15, lh = lane >> 4;
  const int rA = min(strip * 16 + lm, N - 1);
  float* S = &sstd[wave][0];

  v8f acc[8];
#pragma unroll
  for (int t = 0; t < 8; ++t) acc[t] = v8f_zero();
  gemm_panel(h + (size_t)rA * DDIM, Wself, lm, lh, acc);
  gemm_panel(nmean + (size_t)rA * DDIM, Wneigh, lm, lh, acc);

  // standard_output strip -> LDS (per-wave private region; DS ops from the
  // same wave are in-order, so the later reads see these writes)
#pragma unroll
  for (int t = 0; t < 8; ++t) {
    const float bb = bsage[t * 16 + lm];
#pragma unroll
    for (int r = 0; r < 8; ++r)
      S[(r + lh * 8) * 129 + t * 16 + lm] = acc[t][r] + bb;
  }

  // gate GEMM: [std | conv] @ Wg ; std A-fragments come from LDS
  v8f g[8];
#pragma unroll
  for (int t = 0; t < 8; ++t) g[t] = v8f_zero();
  for (int k = 0; k < DDIM; k += 4) {
    const int k0 = k + lh * 2;
    v2f a;
    a.x = S[lm * 129 + k0];
    a.y = S[lm * 129 + k0 + 1];
#pragma unroll
    for (int t = 0; t < 8; ++t) {
      v2f b;
      b.x = Wg[(size_t)k0 * DDIM + t * 16 + lm];
      b.y = Wg[(size_t)(k0 + 1) * DDIM + t * 16 + lm];
      g[t] = wmma_f32(a, b, g[t]);
    }
  }
  gemm_panel(cagg + (size_t)rA * DDIM, Wg + DDIM * DDIM, lm, lh, g);

#pragma unroll
  for (int t = 0; t < 8; ++t) {
    const int col = t * 16 + lm;
    const float bb = bg[col];
#pragma unroll
    for (int r = 0; r < 8; ++r) {
      const int row = r + lh * 8;
      const int gr = strip * 16 + row;
      if (gr < N) {
        const float gate = 1.f / (1.f + expf(-(g[t][r] + bb)));
        const float sv = S[row * 129 + col];
        const float cv = cagg[(size_t)gr * DDIM + col];
        out[(size_t)gr * DDIM + col] = gate * sv + (1.f - gate) * cv;
      }
    }
  }
}

// ----------------------------------------------------------------- launcher
extern "C" void kernel_launch(void* const* d_in, const int* in_sizes, int n_in,
                              void* d_out, int out_size, void* d_ws,
                              size_t ws_size, hipStream_t stream) {
  const float* h = (const float*)d_in[0];
  const float* ctx = (const float*)d_in[1];
  const int* src = (const int*)d_in[2];
  const int* dst = (const int*)d_in[3];
  const float* Wself = (const float*)d_in[4];
  const float* Wneigh = (const float*)d_in[5];
  const float* bsage = (const float*)d_in[6];
  const float* Wm = (const float*)d_in[7];
  const float* bm = (const float*)d_in[8];
  const float* lng = (const float*)d_in[9];
  const float* lnb = (const float*)d_in[10];
  const float* Wg = (const float*)d_in[11];
  const float* bg = (const float*)d_in[12];
  float* out = (float*)d_out;

  const int N = in_sizes[0] / DDIM;
  const int E = in_sizes[2];
  (void)n_in;
  (void)out_size;
  (void)ws_size;

  // workspace carve-up (256B aligned)
  auto alignup = [](size_t x) { return (x + 255) & ~(size_t)255; };
  char* w = (char*)d_ws;
  int* cnt = (int*)w;       w += alignup(sizeof(int) * (size_t)N);
  int* rowptr = (int*)w;    w += alignup(sizeof(int) * (size_t)(N + 1));
  int* cursor = (int*)w;    w += alignup(sizeof(int) * (size_t)N);
  int* part = (int*)w;      w += alignup(sizeof(int) * 4096);
  int* esrc = (int*)w;      w += alignup(sizeof(int) * (size_t)E);
  float* msg = (float*)w;   w += alignup(sizeof(float) * (size_t)N * DDIM);
  float* nmean = (float*)w; w += alignup(sizeof(float) * (size_t)N * DDIM);
  float* cagg = (float*)w;  w += alignup(sizeof(float) * (size_t)N * DDIM);

  const int nStrips = (N + 15) / 16;
  const int nb = (N + SCAN_B - 1) / SCAN_B;

  k_zero<<<(N + 255) / 256, 256, 0, stream>>>(cnt, N);
  k_count<<<512, 256, 0, stream>>>(dst, cnt, E);
  k_scan1<<<nb, SCAN_B, 0, stream>>>(cnt, part, N);
  k_scan2<<<1, 32, 0, stream>>>(part, nb, rowptr + N);
  k_scan3<<<nb, SCAN_B, 0, stream>>>(cnt, part, rowptr, cursor, N);
  k_fill<<<512, 256, 0, stream>>>(src, dst, cursor, esrc, E);

  k_msg<<<(nStrips + 7) / 8, 256, 0, stream>>>(h, ctx, Wm, bm, lng, lnb, msg,
                                               N);
  k_agg<<<(N + 7) / 8, 256, 0, stream>>>(h, msg, rowptr, esrc, nmean, cagg, N);
  k_stdgate<<<(nStrips + 3) / 4, 128, 0, stream>>>(
      h, nmean, cagg, Wself, Wneigh, bsage, Wg, bg, out, N);
}